// GINLayer_12463995093413
// MI455X (gfx1250) — compile-verified
//
#include <hip/hip_runtime.h>

typedef float v2f __attribute__((ext_vector_type(2)));
typedef float v8f __attribute__((ext_vector_type(8)));

#define N_NODES 50000
#define N_EDGES 600000
#define DIM     128
#define EDIM    16
#define BN_EPS  1e-5f

// ---------------------------------------------------------------------------
// Utility: zero a float range
// ---------------------------------------------------------------------------
__global__ void gin_zero_kernel(float* __restrict__ p, int n) {
    int i = blockIdx.x * blockDim.x + threadIdx.x;
    if (i < n) p[i] = 0.0f;
}

// ---------------------------------------------------------------------------
// Edge pass 1: X1 = edge_attr @ ee_w1 + ee_b1  (16x16 tiles, K=16)
// Accumulate per-column sum and sum-of-squares for BN statistics.
// One block = 16 edges; 8 waves, wave w handles output cols [16w,16w+16).
// ---------------------------------------------------------------------------
__global__ void gin_edge_pass1(const float* __restrict__ ea,
                               const float* __restrict__ w1,
                               const float* __restrict__ b1,
                               float* __restrict__ ssum,
                               float* __restrict__ ssq) {
    __shared__ float sA[16][EDIM + 1];
    const int r0  = blockIdx.x * 16;
    const int tid = threadIdx.x;
    { // 256 threads load 16x16 tile of edge_attr
        int i = tid >> 4, j = tid & 15;
        sA[i][j] = ea[(r0 + i) * EDIM + j];
    }
    __syncthreads();

    const int wave = tid >> 5, lane = tid & 31;
    const int half = lane >> 4, l16 = lane & 15;
    const int n = wave * 16 + l16;           // output column 0..127

    v8f acc;
    const float bias = b1[n];
    #pragma unroll
    for (int r = 0; r < 8; ++r) acc[r] = bias;

    #pragma unroll
    for (int k0 = 0; k0 < EDIM; k0 += 4) {
        v2f a, b;
        a.x = sA[l16][k0 + 2 * half];
        a.y = sA[l16][k0 + 2 * half + 1];
        b.x = w1[(k0 + 2 * half) * DIM + n];
        b.y = w1[(k0 + 2 * half + 1) * DIM + n];
        acc = __builtin_amdgcn_wmma_f32_16x16x4_f32(false, a, false, b,
                                                    (short)0, acc, false, false);
    }
    float s1 = 0.f, s2 = 0.f;
    #pragma unroll
    for (int r = 0; r < 8; ++r) { float v = acc[r]; s1 += v; s2 += v * v; }
    atomicAdd(&ssum[n], s1);
    atomicAdd(&ssq[n],  s2);
}

// ---------------------------------------------------------------------------
// BN finalize: scale = gamma * rsqrt(var+eps), shift = beta - mean*scale
// ---------------------------------------------------------------------------
__global__ void gin_bn_finalize(const float* __restrict__ ssum,
                                const float* __restrict__ ssq,
                                const float* __restrict__ gamma,
                                const float* __restrict__ beta,
                                float* __restrict__ scale,
                                float* __restrict__ shift,
                                float inv_count) {
    int c = threadIdx.x;
    if (c < DIM) {
        float mean = ssum[c] * inv_count;
        float var  = ssq[c] * inv_count - mean * mean;
        float sc   = gamma[c] * rsqrtf(var + BN_EPS);
        scale[c] = sc;
        shift[c] = beta[c] - mean * sc;
    }
}

// ---------------------------------------------------------------------------
// Edge pass 2: recompute X1, apply BN+ReLU -> LDS, GEMM with ee_w2 (+b2),
// scatter-atomicAdd rows into agg[dst].
// ---------------------------------------------------------------------------
__global__ void gin_edge_pass2(const float* __restrict__ ea,
                               const long long* __restrict__ eidx,
                               const float* __restrict__ w1,
                               const float* __restrict__ b1,
                               const float* __restrict__ scale0,
                               const float* __restrict__ shift0,
                               const float* __restrict__ w2,
                               const float* __restrict__ b2,
                               float* __restrict__ agg) {
    __shared__ float sA[16][EDIM + 1];
    __shared__ float sY[16][DIM + 4];
    __shared__ int   sDst[16];
    const int r0  = blockIdx.x * 16;
    const int tid = threadIdx.x;
    {
        int i = tid >> 4, j = tid & 15;
        sA[i][j] = ea[(r0 + i) * EDIM + j];
    }
    if (tid < 16) sDst[tid] = (int)eidx[N_EDGES + r0 + tid];
    __syncthreads();

    const int wave = tid >> 5, lane = tid & 31;
    const int half = lane >> 4, l16 = lane & 15;
    const int n = wave * 16 + l16;

    // --- first GEMM: 16 -> 128 ---
    v8f acc;
    const float bias = b1[n];
    #pragma unroll
    for (int r = 0; r < 8; ++r) acc[r] = bias;
    #pragma unroll
    for (int k0 = 0; k0 < EDIM; k0 += 4) {
        v2f a, b;
        a.x = sA[l16][k0 + 2 * half];
        a.y = sA[l16][k0 + 2 * half + 1];
        b.x = w1[(k0 + 2 * half) * DIM + n];
        b.y = w1[(k0 + 2 * half + 1) * DIM + n];
        acc = __builtin_amdgcn_wmma_f32_16x16x4_f32(false, a, false, b,
                                                    (short)0, acc, false, false);
    }
    // --- BN + ReLU -> LDS tile (row = edge-in-tile, col = channel) ---
    const float sc = scale0[n], sh = shift0[n];
    #pragma unroll
    for (int r = 0; r < 8; ++r) {
        float v = acc[r] * sc + sh;
        sY[r + 8 * half][n] = v > 0.f ? v : 0.f;
    }
    __syncthreads();

    // --- second GEMM: 128 -> 128 ---
    v8f acc2;
    const float bias2 = b2[n];
    #pragma unroll
    for (int r = 0; r < 8; ++r) acc2[r] = bias2;
    #pragma unroll
    for (int k0 = 0; k0 < DIM; k0 += 4) {
        v2f a, b;
        a.x = sY[l16][k0 + 2 * half];
        a.y = sY[l16][k0 + 2 * half + 1];
        b.x = w2[(k0 + 2 * half) * DIM + n];
        b.y = w2[(k0 + 2 * half + 1) * DIM + n];
        acc2 = __builtin_amdgcn_wmma_f32_16x16x4_f32(false, a, false, b,
                                                     (short)0, acc2, false, false);
    }
    // --- scatter-add edge embedding into destination node accumulator ---
    #pragma unroll
    for (int r = 0; r < 8; ++r) {
        int er = r + 8 * half;
        atomicAdd(&agg[sDst[er] * DIM + n], acc2[r]);
    }
}

// ---------------------------------------------------------------------------
// hplus = h + agg   (elementwise over N*D)
// ---------------------------------------------------------------------------
__global__ void gin_hplus(const float* __restrict__ h,
                          const float* __restrict__ agg,
                          float* __restrict__ hplus) {
    int i = blockIdx.x * blockDim.x + threadIdx.x;
    if (i < N_NODES * DIM) hplus[i] = h[i] + agg[i];
}

// ---------------------------------------------------------------------------
// neigh[dst] += hplus[src] ; one thread = one (edge, 4-channel chunk)
// ---------------------------------------------------------------------------
__global__ void gin_neigh_scatter(const long long* __restrict__ eidx,
                                  const float* __restrict__ hplus,
                                  float* __restrict__ neigh) {
    int t = blockIdx.x * blockDim.x + threadIdx.x;   // E * 32 threads
    int e = t >> 5;
    int c0 = (t & 31) * 4;
    int s = (int)eidx[e];
    int d = (int)eidx[N_EDGES + e];
    const float4 v = *(const float4*)&hplus[s * DIM + c0];
    float* out = &neigh[d * DIM + c0];
    atomicAdd(out + 0, v.x);
    atomicAdd(out + 1, v.y);
    atomicAdd(out + 2, v.z);
    atomicAdd(out + 3, v.w);
}

// ---------------------------------------------------------------------------
// Node GEMM 1: X = (hplus + neigh) @ mlp_w1 + b1 ; accumulate BN stats.
// ---------------------------------------------------------------------------
__global__ void gin_node_gemm1(const float* __restrict__ hplus,
                               const float* __restrict__ neigh,
                               const float* __restrict__ w1,
                               const float* __restrict__ b1,
                               float* __restrict__ X,
                               float* __restrict__ ssum,
                               float* __restrict__ ssq) {
    __shared__ float sZ[16][DIM + 4];
    const int r0  = blockIdx.x * 16;
    const int tid = threadIdx.x;
    for (int idx = tid; idx < 16 * DIM; idx += 256) {
        int i = idx >> 7, j = idx & 127;
        int off = (r0 + i) * DIM + j;
        sZ[i][j] = hplus[off] + neigh[off];
    }
    __syncthreads();

    const int wave = tid >> 5, lane = tid & 31;
    const int half = lane >> 4, l16 = lane & 15;
    const int n = wave * 16 + l16;

    v8f acc;
    const float bias = b1[n];
    #pragma unroll
    for (int r = 0; r < 8; ++r) acc[r] = bias;
    #pragma unroll
    for (int k0 = 0; k0 < DIM; k0 += 4) {
        v2f a, b;
        a.x = sZ[l16][k0 + 2 * half];
        a.y = sZ[l16][k0 + 2 * half + 1];
        b.x = w1[(k0 + 2 * half) * DIM + n];
        b.y = w1[(k0 + 2 * half + 1) * DIM + n];
        acc = __builtin_amdgcn_wmma_f32_16x16x4_f32(false, a, false, b,
                                                    (short)0, acc, false, false);
    }
    float s1 = 0.f, s2 = 0.f;
    #pragma unroll
    for (int r = 0; r < 8; ++r) {
        float v = acc[r];
        X[(r0 + r + 8 * half) * DIM + n] = v;
        s1 += v; s2 += v * v;
    }
    atomicAdd(&ssum[n], s1);
    atomicAdd(&ssq[n],  s2);
}

// ---------------------------------------------------------------------------
// Node GEMM 2: X2 = relu(bn1(X)) @ mlp_w2 + b2 ; accumulate BN stats.
// ---------------------------------------------------------------------------
__global__ void gin_node_gemm2(const float* __restrict__ X,
                               const float* __restrict__ scale1,
                               const float* __restrict__ shift1,
                               const float* __restrict__ w2,
                               const float* __restrict__ b2,
                               float* __restrict__ X2,
                               float* __restrict__ ssum,
                               float* __restrict__ ssq) {
    __shared__ float sZ[16][DIM + 4];
    const int r0  = blockIdx.x * 16;
    const int tid = threadIdx.x;
    for (int idx = tid; idx < 16 * DIM; idx += 256) {
        int i = idx >> 7, j = idx & 127;
        float v = X[(r0 + i) * DIM + j] * scale1[j] + shift1[j];
        sZ[i][j] = v > 0.f ? v : 0.f;
    }
    __syncthreads();

    const int wave = tid >> 5, lane = tid & 31;
    const int half = lane >> 4, l16 = lane & 15;
    const int n = wave * 16 + l16;

    v8f acc;
    const float bias = b2[n];
    #pragma unroll
    for (int r = 0; r < 8; ++r) acc[r] = bias;
    #pragma unroll
    for (int k0 = 0; k0 < DIM; k0 += 4) {
        v2f a, b;
        a.x = sZ[l16][k0 + 2 * half];
        a.y = sZ[l16][k0 + 2 * half + 1];
        b.x = w2[(k0 + 2 * half) * DIM + n];
        b.y = w2[(k0 + 2 * half + 1) * DIM + n];
        acc = __builtin_amdgcn_wmma_f32_16x16x4_f32(false, a, false, b,
                                                    (short)0, acc, false, false);
    }
    float s1 = 0.f, s2 = 0.f;
    #pragma unroll
    for (int r = 0; r < 8; ++r) {
        float v = acc[r];
        X2[(r0 + r + 8 * half) * DIM + n] = v;
        s1 += v; s2 += v * v;
    }
    atomicAdd(&ssum[n], s1);
    atomicAdd(&ssq[n],  s2);
}

// ---------------------------------------------------------------------------
// Final: out = relu(bn2(X2))
// ---------------------------------------------------------------------------
__global__ void gin_final(const float* __restrict__ X2,
                          const float* __restrict__ scale2,
                          const float* __restrict__ shift2,
                          float* __restrict__ out) {
    int i = blockIdx.x * blockDim.x + threadIdx.x;
    if (i < N_NODES * DIM) {
        int c = i & 127;
        float v = X2[i] * scale2[c] + shift2[c];
        out[i] = v > 0.f ? v : 0.f;
    }
}

// ---------------------------------------------------------------------------
extern "C" void kernel_launch(void* const* d_in, const int* in_sizes, int n_in,
                              void* d_out, int out_size, void* d_ws, size_t ws_size,
                              hipStream_t stream) {
    const float*     h      = (const float*)d_in[0];
    const long long* eidx   = (const long long*)d_in[1];
    const float*     eattr  = (const float*)d_in[2];
    const float*     ee_w1  = (const float*)d_in[3];
    const float*     ee_b1  = (const float*)d_in[4];
    const float*     ee_g1  = (const float*)d_in[5];
    const float*     ee_bb1 = (const float*)d_in[6];
    const float*     ee_w2  = (const float*)d_in[7];
    const float*     ee_b2  = (const float*)d_in[8];
    const float*     mlp_w1 = (const float*)d_in[9];
    const float*     mlp_b1 = (const float*)d_in[10];
    const float*     mlp_g1 = (const float*)d_in[11];
    const float*     mlp_bb1= (const float*)d_in[12];
    const float*     mlp_w2 = (const float*)d_in[13];
    const float*     mlp_b2 = (const float*)d_in[14];
    const float*     mlp_g2 = (const float*)d_in[15];
    const float*     mlp_bb2= (const float*)d_in[16];
    float* out = (float*)d_out;

    const size_t ND = (size_t)N_NODES * DIM;
    float* buf0 = (float*)d_ws;        // agg (edge phase), then X (mlp pre-bn1)
    float* buf1 = buf0 + ND;           // hplus
    float* buf2 = buf1 + ND;           // neigh, then X2
    float* st   = buf2 + ND;           // BN stats: 3 sets x {sum,sq,scale,shift}
    float* sum0 = st;        float* sq0 = st + 128;
    float* scl0 = st + 256;  float* shf0 = st + 384;
    float* sum1 = st + 512;  float* sq1 = st + 640;
    float* scl1 = st + 768;  float* shf1 = st + 896;
    float* sum2 = st + 1024; float* sq2 = st + 1152;
    float* scl2 = st + 1280; float* shf2 = st + 1408;

    // 1) zero accumulators: buf0 (agg) and [buf2 .. st+1536) (neigh + stats)
    {
        int n0 = (int)ND;
        gin_zero_kernel<<<(n0 + 255) / 256, 256, 0, stream>>>(buf0, n0);
        int n1 = (int)ND + 1536;
        gin_zero_kernel<<<(n1 + 255) / 256, 256, 0, stream>>>(buf2, n1);
    }

    // 2) edge encoder stats pass
    gin_edge_pass1<<<N_EDGES / 16, 256, 0, stream>>>(eattr, ee_w1, ee_b1, sum0, sq0);
    gin_bn_finalize<<<1, 128, 0, stream>>>(sum0, sq0, ee_g1, ee_bb1, scl0, shf0,
                                           1.0f / (float)N_EDGES);
    // 3) edge encoder apply + second GEMM + scatter into agg
    gin_edge_pass2<<<N_EDGES / 16, 256, 0, stream>>>(eattr, eidx, ee_w1, ee_b1,
                                                     scl0, shf0, ee_w2, ee_b2, buf0);
    // 4) hplus = h + agg
    gin_hplus<<<(int)(ND + 255) / 256, 256, 0, stream>>>(h, buf0, buf1);
    // 5) neigh[dst] += hplus[src]
    gin_neigh_scatter<<<(N_EDGES * 32) / 256, 256, 0, stream>>>(eidx, buf1, buf2);
    // 6) MLP layer 1 (stats), finalize, layer 2 (stats), finalize
    gin_node_gemm1<<<N_NODES / 16, 256, 0, stream>>>(buf1, buf2, mlp_w1, mlp_b1,
                                                     buf0, sum1, sq1);
    gin_bn_finalize<<<1, 128, 0, stream>>>(sum1, sq1, mlp_g1, mlp_bb1, scl1, shf1,
                                           1.0f / (float)N_NODES);
    gin_node_gemm2<<<N_NODES / 16, 256, 0, stream>>>(buf0, scl1, shf1, mlp_w2, mlp_b2,
                                                     buf2, sum2, sq2);
    gin_bn_finalize<<<1, 128, 0, stream>>>(sum2, sq2, mlp_g2, mlp_bb2, scl2, shf2,
                                           1.0f / (float)N_NODES);
    // 7) final BN + ReLU -> out
    gin_final<<<(int)(ND + 255) / 256, 256, 0, stream>>>(buf2, scl2, shf2, out);
}